// HalfProjCRDLoss_15685220565757
// MI455X (gfx1250) — compile-verified
//
#include <hip/hip_runtime.h>
#include <math.h>

typedef __attribute__((ext_vector_type(2))) float v2f;
typedef __attribute__((ext_vector_type(8))) float v8f;

#define B_SZ    256
#define F_DIM   128
#define K_NEG   4096
#define NCAND   4097           // 1 positive + 4096 negatives
#define CAP_F   100000.0f
#define T_INV   0.02f          // 1/50
#define RESID   0.04096f       // K/CAPACITY
#define EPS_C   1e-7f

// ---- workspace layout (floats) ----
// ws[0..1]   : sum of exp(-d0/T) per j  (for Z[j])
// ws[2..5]   : loss accumulators, index 2 + i*2 + j
// ws[16..)   : f_emb0 (256*128), f_emb1 (256*128)
// ws[WS_E..) : (fast path only) stored e values, plane per i, [j][b][n]
#define WS_F0   16
#define WS_F1   (WS_F0 + B_SZ * F_DIM)
#define WS_E0   (WS_F1 + B_SZ * F_DIM)
#define PLANE   (2 * B_SZ * NCAND)          // one i-plane covering both j
#define WS_E1   (WS_E0 + PLANE)
#define WS_TOTAL (WS_E1 + PLANE)

__global__ __launch_bounds__(32) void init_ws(float* ws) {
    if (threadIdx.x < 6) ws[threadIdx.x] = 0.0f;
}

// ---------------------------------------------------------------------------
// Projection: f_emb = fs @ W^T + b via V_WMMA_F32_16X16X4_F32.
// One wave per 16x16 output tile; K unrolled by 16 (4 WMMAs / iter) so the
// 8 global_load_b64 of the next step overlap the tensor-pipe work.
// ---------------------------------------------------------------------------
__global__ __launch_bounds__(32) void proj_wmma(
    const float* __restrict__ fs0, const float* __restrict__ fs1,
    const float* __restrict__ W0,  const float* __restrict__ b0,
    const float* __restrict__ W1,  const float* __restrict__ b1,
    float* __restrict__ ws)
{
    const int i      = blockIdx.z;
    const float* fs  = i ? fs1 : fs0;
    const float* W   = i ? W1  : W0;
    const float* bia = i ? b1  : b0;
    const int   K    = i ? 1024 : 2048;
    float* out       = ws + (i ? WS_F1 : WS_F0);

    const int lane   = threadIdx.x;
    const int h      = lane >> 4;     // which half-wave
    const int lm     = lane & 15;
    const int m_base = blockIdx.y * 16;
    const int n_base = blockIdx.x * 16;

    const float* arow = fs + (size_t)(m_base + lm) * K + 2 * h;
    const float* brow = W  + (size_t)(n_base + lm) * K + 2 * h;

    v8f c = {};
    for (int k = 0; k < K; k += 16) {
        v2f a0, a1, a2, a3, b0v, b1v, b2v, b3v;
        a0[0] = arow[k     ]; a0[1] = arow[k +  1];
        a1[0] = arow[k +  4]; a1[1] = arow[k +  5];
        a2[0] = arow[k +  8]; a2[1] = arow[k +  9];
        a3[0] = arow[k + 12]; a3[1] = arow[k + 13];
        b0v[0] = brow[k     ]; b0v[1] = brow[k +  1];
        b1v[0] = brow[k +  4]; b1v[1] = brow[k +  5];
        b2v[0] = brow[k +  8]; b2v[1] = brow[k +  9];
        b3v[0] = brow[k + 12]; b3v[1] = brow[k + 13];
        c = __builtin_amdgcn_wmma_f32_16x16x4_f32(false, a0, false, b0v, (short)0, c, false, false);
        c = __builtin_amdgcn_wmma_f32_16x16x4_f32(false, a1, false, b1v, (short)0, c, false, false);
        c = __builtin_amdgcn_wmma_f32_16x16x4_f32(false, a2, false, b2v, (short)0, c, false, false);
        c = __builtin_amdgcn_wmma_f32_16x16x4_f32(false, a3, false, b3v, (short)0, c, false, false);
    }

    const float bv = bia[n_base + lm];
    #pragma unroll
    for (int v = 0; v < 8; ++v) {
        const int row = m_base + v + 8 * h;
        out[(size_t)row * F_DIM + n_base + lm] = c[v] + bv;
    }
}

// ---------------------------------------------------------------------------
// Block reduction + global accumulate
// ---------------------------------------------------------------------------
__device__ inline void block_reduce_add(float v, float* red, float* acc) {
    const int t = threadIdx.x;
    red[t] = v;
    __syncthreads();
    for (int s = 128; s > 0; s >>= 1) {
        if (t < s) red[t] += red[t + s];
        __syncthreads();
    }
    if (t == 0) atomicAdd(acc, red[0]);
    __syncthreads();
}

// Gather one memory row and compute both squared distances.
__device__ inline void dist2_both(const float* __restrict__ w,
                                  const float* __restrict__ f0,
                                  const float* __restrict__ f1,
                                  float& d20, float& d21)
{
    // Prefetch the remaining cache lines of this 512B row (global_prefetch_b8).
    __builtin_prefetch((const void*)(w + 32), 0, 0);
    __builtin_prefetch((const void*)(w + 64), 0, 0);
    __builtin_prefetch((const void*)(w + 96), 0, 0);
    d20 = 0.0f; d21 = 0.0f;
    #pragma unroll 4
    for (int k = 0; k < F_DIM; k += 4) {
        const float4 wv = *(const float4*)(w + k);
        float a, c;
        a = f0[k]     - wv.x; d20 += a * a;  c = f1[k]     - wv.x; d21 += c * c;
        a = f0[k + 1] - wv.y; d20 += a * a;  c = f1[k + 1] - wv.y; d21 += c * c;
        a = f0[k + 2] - wv.z; d20 += a * a;  c = f1[k + 2] - wv.z; d21 += c * c;
        a = f0[k + 3] - wv.w; d20 += a * a;  c = f1[k + 3] - wv.w; d21 += c * c;
    }
}

// ---------------------------------------------------------------------------
// FAST PATH pass 1: single gather pass. Computes e0,e1 for every (j,b,n),
// stores them to ws, and accumulates sum(e0) per j (-> Z[j]).
// ---------------------------------------------------------------------------
__global__ __launch_bounds__(256) void gather_pass_store(
    const float* __restrict__ memory, const int* __restrict__ idx,
    const int* __restrict__ cidx, float* __restrict__ ws)
{
    const int j = blockIdx.z;
    const int b = blockIdx.y;
    const int n = blockIdx.x * 256 + threadIdx.x;

    __shared__ float f0[F_DIM];
    __shared__ float f1[F_DIM];
    __shared__ float red[256];

    if (threadIdx.x < F_DIM) {
        f0[threadIdx.x] = ws[WS_F0 + (size_t)b * F_DIM + threadIdx.x];
        f1[threadIdx.x] = ws[WS_F1 + (size_t)b * F_DIM + threadIdx.x];
    }
    __syncthreads();

    float e0 = 0.0f, e1 = 0.0f;
    if (n < NCAND) {
        const int widx = (n == 0) ? idx[b] : cidx[(size_t)b * K_NEG + (n - 1)];
        const float* w = memory + ((size_t)j * 100000 + (size_t)widx) * F_DIM;
        float d20, d21;
        dist2_both(w, f0, f1, d20, d21);
        e0 = expf(-sqrtf(d20) * T_INV);
        e1 = expf(-sqrtf(d21) * T_INV);
        const size_t off = ((size_t)j * B_SZ + b) * NCAND + n;
        ws[WS_E0 + off] = e0;
        ws[WS_E1 + off] = e1;
    }
    block_reduce_add(e0, red, &ws[j]);
}

// FAST PATH pass 2: pure reduction over stored e values.
__global__ __launch_bounds__(256) void loss_from_store(float* __restrict__ ws)
{
    const int j = blockIdx.z;
    const int b = blockIdx.y;
    const int n = blockIdx.x * 256 + threadIdx.x;

    __shared__ float red[256];

    const float Z    = ws[j] * (CAP_F / ((float)B_SZ * (float)NCAND));
    const float Zinv = 1.0f / Z;

    float t0 = 0.0f, t1 = 0.0f;
    if (n < NCAND) {
        const size_t off = ((size_t)j * B_SZ + b) * NCAND + n;
        const float p0 = ws[WS_E0 + off] * Zinv;
        const float p1 = ws[WS_E1 + off] * Zinv;
        if (n == 0) {
            t0 = logf(p0 / (p0 + RESID + EPS_C));
            t1 = logf(p1 / (p1 + RESID + EPS_C));
        } else {
            t0 = logf(RESID / (p0 + RESID + EPS_C));
            t1 = logf(RESID / (p1 + RESID + EPS_C));
        }
    }
    block_reduce_add(t0, red, &ws[2 + 0 * 2 + j]);   // i = 0
    block_reduce_add(t1, red, &ws[2 + 1 * 2 + j]);   // i = 1
}

// ---------------------------------------------------------------------------
// FALLBACK (small ws): original recompute scheme.
// ---------------------------------------------------------------------------
__global__ __launch_bounds__(256) void z_pass(
    const float* __restrict__ memory, const int* __restrict__ idx,
    const int* __restrict__ cidx, float* __restrict__ ws)
{
    const int j = blockIdx.z;
    const int b = blockIdx.y;
    const int n = blockIdx.x * 256 + threadIdx.x;

    __shared__ float f0[F_DIM];
    __shared__ float red[256];

    if (threadIdx.x < F_DIM)
        f0[threadIdx.x] = ws[WS_F0 + (size_t)b * F_DIM + threadIdx.x];
    __syncthreads();

    float e0 = 0.0f;
    if (n < NCAND) {
        const int widx = (n == 0) ? idx[b] : cidx[(size_t)b * K_NEG + (n - 1)];
        const float* w = memory + ((size_t)j * 100000 + (size_t)widx) * F_DIM;
        float d2 = 0.0f;
        #pragma unroll 4
        for (int k = 0; k < F_DIM; k += 4) {
            const float4 wv = *(const float4*)(w + k);
            const float t0 = f0[k]     - wv.x;
            const float t1 = f0[k + 1] - wv.y;
            const float t2 = f0[k + 2] - wv.z;
            const float t3 = f0[k + 3] - wv.w;
            d2 += t0 * t0 + t1 * t1 + t2 * t2 + t3 * t3;
        }
        e0 = expf(-sqrtf(d2) * T_INV);
    }
    block_reduce_add(e0, red, &ws[j]);
}

__global__ __launch_bounds__(256) void loss_pass(
    const float* __restrict__ memory, const int* __restrict__ idx,
    const int* __restrict__ cidx, float* __restrict__ ws)
{
    const int j = blockIdx.z;
    const int b = blockIdx.y;
    const int n = blockIdx.x * 256 + threadIdx.x;

    __shared__ float f0[F_DIM];
    __shared__ float f1[F_DIM];
    __shared__ float red[256];

    if (threadIdx.x < F_DIM) {
        f0[threadIdx.x] = ws[WS_F0 + (size_t)b * F_DIM + threadIdx.x];
        f1[threadIdx.x] = ws[WS_F1 + (size_t)b * F_DIM + threadIdx.x];
    }
    __syncthreads();

    const float Z    = ws[j] * (CAP_F / ((float)B_SZ * (float)NCAND));
    const float Zinv = 1.0f / Z;

    float t0 = 0.0f, t1 = 0.0f;
    if (n < NCAND) {
        const int widx = (n == 0) ? idx[b] : cidx[(size_t)b * K_NEG + (n - 1)];
        const float* w = memory + ((size_t)j * 100000 + (size_t)widx) * F_DIM;
        float d20, d21;
        dist2_both(w, f0, f1, d20, d21);
        const float p0 = expf(-sqrtf(d20) * T_INV) * Zinv;
        const float p1 = expf(-sqrtf(d21) * T_INV) * Zinv;
        if (n == 0) {
            t0 = logf(p0 / (p0 + RESID + EPS_C));
            t1 = logf(p1 / (p1 + RESID + EPS_C));
        } else {
            t0 = logf(RESID / (p0 + RESID + EPS_C));
            t1 = logf(RESID / (p1 + RESID + EPS_C));
        }
    }
    block_reduce_add(t0, red, &ws[2 + 0 * 2 + j]);   // i = 0
    block_reduce_add(t1, red, &ws[2 + 1 * 2 + j]);   // i = 1
}

__global__ __launch_bounds__(32) void finalize(const float* __restrict__ ws,
                                               float* __restrict__ out)
{
    if (threadIdx.x == 0) {
        const float l00 = -ws[2] / (float)B_SZ;
        const float l01 = -ws[3] / (float)B_SZ;
        const float l10 = -ws[4] / (float)B_SZ;
        const float l11 = -ws[5] / (float)B_SZ;
        out[0] = l00 + l01 + l10 + l11;
        out[1] = l00;
        out[2] = l01;
        out[3] = l10;
        out[4] = l11;
    }
}

extern "C" void kernel_launch(void* const* d_in, const int* in_sizes, int n_in,
                              void* d_out, int out_size, void* d_ws, size_t ws_size,
                              hipStream_t stream) {
    const float* fs0  = (const float*)d_in[0];
    const float* fs1  = (const float*)d_in[1];
    const float* W0   = (const float*)d_in[2];
    const float* b0   = (const float*)d_in[3];
    const float* W1   = (const float*)d_in[4];
    const float* b1   = (const float*)d_in[5];
    const float* mem  = (const float*)d_in[6];
    const int*   idx  = (const int*)d_in[7];
    const int*   cidx = (const int*)d_in[8];
    float* ws  = (float*)d_ws;
    float* out = (float*)d_out;

    init_ws<<<1, 32, 0, stream>>>(ws);

    // 16x16 tiles: N=128/16=8, M=256/16=16, two projections (grid.z)
    proj_wmma<<<dim3(8, 16, 2), 32, 0, stream>>>(fs0, fs1, W0, b0, W1, b1, ws);

    const int nchunks = (NCAND + 255) / 256;   // 17
    const dim3 grid(nchunks, B_SZ, 2);

    if (ws_size >= (size_t)WS_TOTAL * sizeof(float)) {
        // single gather pass + cheap reduction pass (halves L2 gather traffic)
        gather_pass_store<<<grid, 256, 0, stream>>>(mem, idx, cidx, ws);
        loss_from_store<<<grid, 256, 0, stream>>>(ws);
    } else {
        z_pass<<<grid, 256, 0, stream>>>(mem, idx, cidx, ws);
        loss_pass<<<grid, 256, 0, stream>>>(mem, idx, cidx, ws);
    }

    finalize<<<1, 32, 0, stream>>>(ws, out);
}